// Block_73083163508880
// MI455X (gfx1250) — compile-verified
//
#include <hip/hip_runtime.h>

#define NN 100000   // nodes per type
#define NE 2000000  // edges per relation
#define H  64       // hidden channels

typedef __attribute__((ext_vector_type(2))) float v2f;
typedef __attribute__((ext_vector_type(8))) float v8f;

__device__ __forceinline__ v8f wmma4(v2f a, v2f b, v8f c) {
  // D = A(16x4 f32) * B(4x16 f32) + C(16x16 f32)
  return __builtin_amdgcn_wmma_f32_16x16x4_f32(false, a, false, b, (short)0, c,
                                               false, false);
}

// ---------------------------------------------------------------- utilities
__global__ void __launch_bounds__(256) fill_f32(float* p, float v, long n) {
  long i = (long)blockIdx.x * blockDim.x + threadIdx.x;
  if (i < n) p[i] = v;
}

// ---------------------------------------------------------------- SAGE scatter
// one thread per (edge, 16-channel quarter); all SAGE sources are 64-channel
__global__ void __launch_bounds__(256)
sage_scatter(const float* __restrict__ xsrc, const int* __restrict__ src,
             const int* __restrict__ dst, float* __restrict__ agg,
             float* __restrict__ cnt) {
  long t = (long)blockIdx.x * blockDim.x + threadIdx.x;
  if (t >= (long)NE * 4) return;
  int e = (int)(t >> 2), q = (int)(t & 3);
  int s = src[e], d = dst[e];
  const float4* xp = (const float4*)(xsrc + (long)s * H) + q * 4;
  float* ap = agg + (long)d * H + q * 16;
#pragma unroll
  for (int j = 0; j < 4; ++j) {
    float4 v = xp[j];
    atomicAdd(ap + 4 * j + 0, v.x);
    atomicAdd(ap + 4 * j + 1, v.y);
    atomicAdd(ap + 4 * j + 2, v.z);
    atomicAdd(ap + 4 * j + 3, v.w);
  }
  if (q == 0) atomicAdd(cnt + d, 1.0f);
}

// ---------------------------------------------------------------- SAGE finish
// out = L2norm( (agg/max(cnt,1))@Wl + b + xdst@Wr ), then ReLU.
// 128 threads = 4 waves; block handles 16 rows; wave w does cols [16w,16w+16).
__global__ void __launch_bounds__(128)
sage_finish(const float* __restrict__ agg, const float* __restrict__ cnt,
            const float* __restrict__ xdst, int Cd,
            const float* __restrict__ Wl, const float* __restrict__ bias,
            const float* __restrict__ Wr, float* __restrict__ out) {
  __shared__ float xs[16][H];      // mean-aggregated source features
  __shared__ float xd[16][H];      // destination features (Cd cols used)
  __shared__ float os[16][H];      // output tile
  __shared__ float scl[16];
  int row0 = blockIdx.x * 16;
  int tid = threadIdx.x;
  for (int i = tid; i < 16 * H; i += 128) {
    int r = i >> 6, c = i & 63;
    float cc = cnt[row0 + r];
    xs[r][c] = agg[(long)(row0 + r) * H + c] / fmaxf(cc, 1.0f);
  }
  for (int i = tid; i < 16 * Cd; i += 128) {
    int r = i / Cd, c = i % Cd;
    xd[r][c] = xdst[(long)(row0 + r) * Cd + c];
  }
  __syncthreads();
  int wave = tid >> 5, lane = tid & 31;
  int m = lane & 15;
  int kh = (lane >> 4) * 2;  // 0 or 2: which K pair this half-wave carries
  int col = wave * 16 + m;
  v8f acc = {};
  for (int k0 = 0; k0 < H; k0 += 4) {
    v2f a; a.x = xs[m][k0 + kh];            a.y = xs[m][k0 + kh + 1];
    v2f b; b.x = Wl[(k0 + kh) * H + col];   b.y = Wl[(k0 + kh + 1) * H + col];
    acc = wmma4(a, b, acc);
  }
  for (int k0 = 0; k0 < Cd; k0 += 4) {
    v2f a; a.x = xd[m][k0 + kh];            a.y = xd[m][k0 + kh + 1];
    v2f b; b.x = Wr[(k0 + kh) * H + col];   b.y = Wr[(k0 + kh + 1) * H + col];
    acc = wmma4(a, b, acc);
  }
  float bcol = bias[col];
  int rb = (lane >> 4) * 8;
#pragma unroll
  for (int r = 0; r < 8; ++r) os[rb + r][col] = acc[r] + bcol;
  __syncthreads();
  if (tid < 16) {
    float ss = 0.f;
    for (int c = 0; c < H; ++c) { float v = os[tid][c]; ss += v * v; }
    scl[tid] = 1.0f / fmaxf(sqrtf(ss), 1e-12f);
  }
  __syncthreads();
  for (int i = tid; i < 16 * H; i += 128) {
    int r = i >> 6, c = i & 63;
    out[(long)(row0 + r) * H + c] = fmaxf(os[r][c] * scl[r], 0.0f);
  }
}

// ---------------------------------------------------------------- GEMM [N,64]x[64,64]
__global__ void __launch_bounds__(128)
gemm_n64(const float* __restrict__ X, const float* __restrict__ W,
         float* __restrict__ Y) {
  __shared__ float xs[16][H];
  int row0 = blockIdx.x * 16;
  for (int i = threadIdx.x; i < 16 * H; i += 128)
    xs[i >> 6][i & 63] = X[(long)(row0 + (i >> 6)) * H + (i & 63)];
  __syncthreads();
  int wave = threadIdx.x >> 5, lane = threadIdx.x & 31;
  int m = lane & 15, kh = (lane >> 4) * 2, col = wave * 16 + m;
  v8f acc = {};
  for (int k0 = 0; k0 < H; k0 += 4) {
    v2f a; a.x = xs[m][k0 + kh];          a.y = xs[m][k0 + kh + 1];
    v2f b; b.x = W[(k0 + kh) * H + col];  b.y = W[(k0 + kh + 1) * H + col];
    acc = wmma4(a, b, acc);
  }
  int rb = (lane >> 4) * 8;
#pragma unroll
  for (int r = 0; r < 8; ++r) Y[(long)(row0 + rb + r) * H + col] = acc[r];
}

// ---------------------------------------------------------------- GCN pieces
__global__ void __launch_bounds__(256)
count_edges(const int* __restrict__ dst, float* __restrict__ deg) {
  long e = (long)blockIdx.x * blockDim.x + threadIdx.x;
  if (e < NE) atomicAdd(deg + dst[e], 1.0f);
}

__global__ void __launch_bounds__(256) to_dinv(float* deg) {
  long i = (long)blockIdx.x * blockDim.x + threadIdx.x;
  if (i < NN) deg[i] = rsqrtf(deg[i] + 1.0f);  // self loop adds 1
}

__global__ void __launch_bounds__(256)
gcn_scatter(const float* __restrict__ xw, const int* __restrict__ src,
            const int* __restrict__ dst, const float* __restrict__ dinv,
            float* __restrict__ acc) {
  long t = (long)blockIdx.x * blockDim.x + threadIdx.x;
  if (t >= (long)NE * 4) return;
  int e = (int)(t >> 2), q = (int)(t & 3);
  int s = src[e], d = dst[e];
  float coef = dinv[s] * dinv[d];
  const float4* xp = (const float4*)(xw + (long)s * H) + q * 4;
  float* ap = acc + (long)d * H + q * 16;
#pragma unroll
  for (int j = 0; j < 4; ++j) {
    float4 v = xp[j];
    atomicAdd(ap + 4 * j + 0, v.x * coef);
    atomicAdd(ap + 4 * j + 1, v.y * coef);
    atomicAdd(ap + 4 * j + 2, v.z * coef);
    atomicAdd(ap + 4 * j + 3, v.w * coef);
  }
}

__global__ void __launch_bounds__(256)
gcn_finish(const float* __restrict__ acc, const float* __restrict__ xw,
           const float* __restrict__ dinv, const float* __restrict__ bias,
           float* __restrict__ out) {
  long i = (long)blockIdx.x * blockDim.x + threadIdx.x;
  if (i >= (long)NN * H) return;
  int c = (int)(i & 63);
  int r = (int)(i >> 6);
  float di = dinv[r];
  out[i] = fmaxf(acc[i] + xw[i] * di * di + bias[c], 0.0f);
}

// ---------------------------------------------------------------- BatchNorm
__global__ void __launch_bounds__(256)
bn_reduce(const float* __restrict__ x, float* __restrict__ stats) {
  __shared__ float ssum[256], ssq[256];
  int c = threadIdx.x & 63, sub = threadIdx.x >> 6;
  long rend = (long)(blockIdx.x + 1) * 1024;
  if (rend > NN) rend = NN;
  float s = 0.f, q = 0.f;
  for (long r = (long)blockIdx.x * 1024 + sub; r < rend; r += 4) {
    float v = x[r * H + c];
    s += v; q += v * v;
  }
  ssum[threadIdx.x] = s; ssq[threadIdx.x] = q;
  __syncthreads();
  if (sub == 0) {
    s = ssum[c] + ssum[64 + c] + ssum[128 + c] + ssum[192 + c];
    q = ssq[c] + ssq[64 + c] + ssq[128 + c] + ssq[192 + c];
    atomicAdd(stats + c, s);
    atomicAdd(stats + 64 + c, q);
  }
}

__global__ void __launch_bounds__(256)
bn_apply(float* __restrict__ x, const float* __restrict__ stats,
         const float* __restrict__ g, const float* __restrict__ b) {
  long i = (long)blockIdx.x * blockDim.x + threadIdx.x;
  if (i >= (long)NN * H) return;
  int c = (int)(i & 63);
  float inv_n = 1.0f / (float)NN;
  float mean = stats[c] * inv_n;
  float var = stats[64 + c] * inv_n - mean * mean;
  x[i] = (x[i] - mean) * rsqrtf(var + 1e-5f) * g[c] + b[c];
}

// ---------------------------------------------------------------- host side
static inline int cdiv_l(long a, int b) { return (int)((a + b - 1) / b); }

static void run_sage(const float* xsrc, const int* ei, const float* xdst, int Cd,
                     const float* Wl, const float* b, const float* Wr,
                     float* out, float* agg, float* cnt, hipStream_t stream) {
  fill_f32<<<cdiv_l((long)NN * H, 256), 256, 0, stream>>>(agg, 0.f, (long)NN * H);
  fill_f32<<<cdiv_l(NN, 256), 256, 0, stream>>>(cnt, 0.f, NN);
  sage_scatter<<<cdiv_l((long)NE * 4, 256), 256, 0, stream>>>(xsrc, ei, ei + NE,
                                                              agg, cnt);
  sage_finish<<<NN / 16, 128, 0, stream>>>(agg, cnt, xdst, Cd, Wl, b, Wr, out);
}

static void run_gcn_bn(const float* x, const int* ei, const float* W,
                       const float* b, const float* gamma, const float* beta,
                       float* out, float* xw, float* acc, float* deg,
                       float* stats, hipStream_t stream) {
  gemm_n64<<<NN / 16, 128, 0, stream>>>(x, W, xw);
  fill_f32<<<cdiv_l(NN, 256), 256, 0, stream>>>(deg, 0.f, NN);
  count_edges<<<cdiv_l(NE, 256), 256, 0, stream>>>(ei + NE, deg);
  to_dinv<<<cdiv_l(NN, 256), 256, 0, stream>>>(deg);
  fill_f32<<<cdiv_l((long)NN * H, 256), 256, 0, stream>>>(acc, 0.f, (long)NN * H);
  gcn_scatter<<<cdiv_l((long)NE * 4, 256), 256, 0, stream>>>(xw, ei, ei + NE,
                                                             deg, acc);
  gcn_finish<<<cdiv_l((long)NN * H, 256), 256, 0, stream>>>(acc, xw, deg, b, out);
  fill_f32<<<1, 256, 0, stream>>>(stats, 0.f, 128);
  bn_reduce<<<cdiv_l(NN, 1024), 256, 0, stream>>>(out, stats);
  bn_apply<<<cdiv_l((long)NN * H, 256), 256, 0, stream>>>(out, stats, gamma, beta);
}

extern "C" void kernel_launch(void* const* d_in, const int* in_sizes, int n_in,
                              void* d_out, int out_size, void* d_ws,
                              size_t ws_size, hipStream_t stream) {
  const float* game_x  = (const float*)d_in[0];
  const float* state_x = (const float*)d_in[1];
  const float* pc_x    = (const float*)d_in[2];
  const int* ei_vv  = (const int*)d_in[3];
  const int* ei_hvs = (const int*)d_in[4];
  const int* ei_hsv = (const int*)d_in[5];
  const int* ei_ivs = (const int*)d_in[6];
  const int* ei_isv = (const int*)d_in[7];
  const int* ei_ss  = (const int*)d_in[8];
  const int* ei_pp  = (const int*)d_in[9];
  const int* ei_ps  = (const int*)d_in[10];
  const int* ei_sp  = (const int*)d_in[11];
  const float* sWl[6]; const float* sb[6]; const float* sWr[6];
  for (int i = 0; i < 6; ++i) {
    sWl[i] = (const float*)d_in[12 + 3 * i];
    sb[i]  = (const float*)d_in[13 + 3 * i];
    sWr[i] = (const float*)d_in[14 + 3 * i];
  }
  const float* gcfg_W = (const float*)d_in[30];
  const float* gcfg_b = (const float*)d_in[31];
  const float* gpc_W  = (const float*)d_in[32];
  const float* gpc_b  = (const float*)d_in[33];
  const float* gst_W  = (const float*)d_in[34];
  const float* gst_b  = (const float*)d_in[35];
  const float* bncfg_g = (const float*)d_in[36];
  const float* bncfg_b = (const float*)d_in[37];
  const float* bnpc_g  = (const float*)d_in[38];
  const float* bnpc_b  = (const float*)d_in[39];
  const float* bnst_g  = (const float*)d_in[40];
  const float* bnst_b  = (const float*)d_in[41];

  float* OUT_S = (float*)d_out;
  float* OUT_G = OUT_S + (long)NN * H;
  float* OUT_P = OUT_G + (long)NN * H;

  float* B0    = (float*)d_ws;          // scatter accumulator
  float* B1    = B0 + (long)NN * H;     // xw scratch for GCN
  float* BG    = B1 + (long)NN * H;     // g / s working buffer
  float* CNT   = BG + (long)NN * H;     // counts / dinv
  float* STATS = CNT + NN;              // 128 floats of BN stats

  // g = relu(sage(state, game, hist_s_v, s1))
  run_sage(state_x, ei_hsv, game_x, 32, sWl[0], sb[0], sWr[0], BG, B0, CNT, stream);
  // g = relu(sage(state, g, in_s_v, s2))   (in place per-row)
  run_sage(state_x, ei_isv, BG, 64, sWl[1], sb[1], sWr[1], BG, B0, CNT, stream);
  // p = relu(sage(state, pc, s_pc, s3))
  run_sage(state_x, ei_sp, pc_x, 32, sWl[2], sb[2], sWr[2], OUT_P, B0, CNT, stream);
  // g = bn(relu(gcn(g, v_v)))
  run_gcn_bn(BG, ei_vv, gcfg_W, gcfg_b, bncfg_g, bncfg_b, OUT_G, B1, B0, CNT,
             STATS, stream);
  // p = bn(relu(gcn(p, pc_pc)))
  run_gcn_bn(OUT_P, ei_pp, gpc_W, gpc_b, bnpc_g, bnpc_b, OUT_P, B1, B0, CNT,
             STATS, stream);
  // s = relu(sage(g, state, hist_v_s, s4))
  run_sage(OUT_G, ei_hvs, state_x, 64, sWl[3], sb[3], sWr[3], BG, B0, CNT, stream);
  // s = relu(sage(g, s, in_v_s, s5))
  run_sage(OUT_G, ei_ivs, BG, 64, sWl[4], sb[4], sWr[4], BG, B0, CNT, stream);
  // s = relu(sage(p, s, pc_s, s6))
  run_sage(OUT_P, ei_ps, BG, 64, sWl[5], sb[5], sWr[5], BG, B0, CNT, stream);
  // s = bn(relu(gcn(s, s_s)))
  run_gcn_bn(BG, ei_ss, gst_W, gst_b, bnst_g, bnst_b, OUT_S, B1, B0, CNT,
             STATS, stream);
}